// GCN_35948876268151
// MI455X (gfx1250) — compile-verified
//
#include <hip/hip_runtime.h>

typedef __attribute__((ext_vector_type(2))) float v2f;
typedef __attribute__((ext_vector_type(8))) float v8f;

#define EMB 256
#define HID 64
#define OUTF 32

#define KCHUNK 128
#define BSTRIDE 72   // LDS row stride (floats): 2*72 mod 64 = 16 -> lane halves hit disjoint banks

static inline int ceil_div_i(long long a, long long b) { return (int)((a + b - 1) / b); }

// ---------------------------------------------------------------------------
// FP32 WMMA GEMM: C[M x Ncols] = act( A[M x K] * B[K x Ncols] + bias )
// One wave computes a 16-row x (NT*16)-col strip: A fragment loaded once,
// reused across NT WMMAs (4x cut of the dominant gathered-A HBM traffic).
// B panel for the block's column chunk is staged into LDS with
// GLOBAL_LOAD_ASYNC_TO_LDS_B128 (ASYNCcnt-tracked direct cache->LDS copy).
// Fragment layouts per CDNA5 ISA 7.12.2 (V_WMMA_F32_16X16X4_F32):
//   A 16x4 : lane<16 -> {K=0,K=1}, lane>=16 -> {K=2,K=3}, M = lane&15
//   B 4x16 : lane<16 -> rows {K=0,K=1}, lane>=16 -> rows {K=2,K=3}, N = lane&15
//   C/D    : VGPR r -> M = r + 8*(lane>>4), N = lane&15
// ---------------------------------------------------------------------------
template <int NT>
__global__ void __launch_bounds__(256)
wmma_gemm_f32_kernel(const float* __restrict__ A,
                     const int* __restrict__ gidx,
                     int lda,
                     const float* __restrict__ Bm,
                     const float* __restrict__ bias,
                     float* __restrict__ C,
                     int M, int Ncols, int K, int relu)
{
    __shared__ float Bs[KCHUNK * BSTRIDE];  // 36 KB

    const int lane    = threadIdx.x & 31;
    const int wv      = threadIdx.x >> 5;        // 8 waves / block
    const int tileM   = blockIdx.x * 8 + wv;     // 16-row strip per wave
    const int colBase = blockIdx.y * (NT * 16);  // column chunk per block
    const int ncols   = NT * 16;

    const int r  = lane & 15;        // A row-in-tile / B col-in-tile
    const int kk = (lane >> 4) << 1; // 0 (lanes 0-15) or 2 (lanes 16-31)

    const bool waveActive = (tileM * 16 < M);    // wave-uniform: EXEC stays full

    int arow = tileM * 16 + r;
    if (arow > M - 1) arow = M - 1;              // clamp tail rows (discarded on store)
    if (waveActive && gidx) arow = gidx[arow];   // fused embedding gather
    const float* __restrict__ Arow = A + (size_t)arow * (size_t)lda;

    v8f acc[NT];
#pragma unroll
    for (int t = 0; t < NT; ++t) acc[t] = (v8f){};

    for (int k0 = 0; k0 < K; k0 += KCHUNK) {
        const int kc = (K - k0 < KCHUNK) ? (K - k0) : KCHUNK;

        // Stage B[k0..k0+kc) x [colBase..colBase+ncols) into LDS, 16B per lane.
        // Trip count is block-uniform (kc*ncols/4 is a multiple of 256).
        const int nvec = (kc * ncols) >> 2;
        for (int i = threadIdx.x; i < nvec; i += 256) {
            const int krow = i / (ncols >> 2);
            const int c4   = (i - krow * (ncols >> 2)) << 2;
            const float* gp = Bm + (size_t)(k0 + krow) * Ncols + colBase + c4;
            unsigned lds = (unsigned)(size_t)(&Bs[krow * BSTRIDE + c4]);
#if defined(__HIP_DEVICE_COMPILE__)
            asm volatile("global_load_async_to_lds_b128 %0, %1, off"
                         :: "v"(lds), "v"(gp) : "memory");
#else
            (void)gp; (void)lds;
#endif
        }
#if defined(__HIP_DEVICE_COMPILE__)
        asm volatile("s_wait_asynccnt 0" ::: "memory");
#endif
        __syncthreads();

        if (waveActive) {
            for (int k = 0; k < kc; k += 4) {
                v2f a;
                a.x = Arow[k0 + k + kk];
                a.y = Arow[k0 + k + kk + 1];
#pragma unroll
                for (int t = 0; t < NT; ++t) {
                    v2f b;
                    b.x = Bs[(k + kk)     * BSTRIDE + t * 16 + r];
                    b.y = Bs[(k + kk + 1) * BSTRIDE + t * 16 + r];
                    acc[t] = __builtin_amdgcn_wmma_f32_16x16x4_f32(
                        /*neg_a=*/false, a, /*neg_b=*/false, b,
                        /*c_mod=*/(short)0, acc[t],
                        /*reuse_a=*/false, /*reuse_b=*/false);
                }
            }
        }
        __syncthreads();
    }

    if (!waveActive) return;

    const int hi = lane >> 4;
#pragma unroll
    for (int t = 0; t < NT; ++t) {
        const int col = colBase + t * 16 + (lane & 15);
        const float bv = bias ? bias[col] : 0.0f;
#pragma unroll
        for (int rr = 0; rr < 8; ++rr) {
            const int row = tileM * 16 + rr + hi * 8;
            if (row < M) {
                float v = acc[t][rr] + bv;
                if (relu) v = fmaxf(v, 0.0f);
                C[(size_t)row * Ncols + col] = v;
            }
        }
    }
}

// ---------------------------------------------------------------------------
// Elementwise / graph kernels
// ---------------------------------------------------------------------------
__global__ void fill_kernel(float* __restrict__ p, float v, long long n) {
    long long i = (long long)blockIdx.x * blockDim.x + threadIdx.x;
    if (i < n) p[i] = v;
}

__global__ void deg_accum_kernel(const int* __restrict__ dst,
                                 const float* __restrict__ ew,
                                 float* __restrict__ deg, int E) {
    int e = blockIdx.x * blockDim.x + threadIdx.x;
    if (e < E) atomicAdd(&deg[dst[e]], ew[e]);
}

__global__ void rsqrt_inplace_kernel(float* __restrict__ deg, int N) {
    int i = blockIdx.x * blockDim.x + threadIdx.x;
    if (i < N) deg[i] = rsqrtf(deg[i]);
}

__global__ void norm_kernel(const int* __restrict__ src, const int* __restrict__ dst,
                            const float* __restrict__ ew, const float* __restrict__ dinv,
                            float* __restrict__ nrm, int E) {
    int e = blockIdx.x * blockDim.x + threadIdx.x;
    if (e < E) nrm[e] = dinv[src[e]] * ew[e] * dinv[dst[e]];
}

// 16 consecutive threads per edge, 4 features each: b128 row reads of
// hw[src[e]] and a coalesced stream of f32 atomics into agg[dst[e]] (L2).
__global__ void edge_scatter_kernel(const int* __restrict__ src,
                                    const int* __restrict__ dst,
                                    const float* __restrict__ nrm,
                                    const float* __restrict__ hw,
                                    float* __restrict__ agg,
                                    long long total) {  // total = E*16
    long long t = (long long)blockIdx.x * blockDim.x + threadIdx.x;
    if (t >= total) return;
    const int e = (int)(t >> 4);
    const int q = (int)(t & 15) << 2;
    const float w = nrm[e];
    const float4 hv = *(const float4*)(hw + (size_t)src[e] * HID + q);
    float* ap = agg + (size_t)dst[e] * HID + q;
    atomicAdd(ap + 0, w * hv.x);
    atomicAdd(ap + 1, w * hv.y);
    atomicAdd(ap + 2, w * hv.z);
    atomicAdd(ap + 3, w * hv.w);
}

__global__ void combine_kernel(const float4* __restrict__ agg,
                               const float4* __restrict__ hw,
                               const float* __restrict__ dinv,
                               const float4* __restrict__ bias,
                               float4* __restrict__ hout, int N, int relu) {
    long long i = (long long)blockIdx.x * blockDim.x + threadIdx.x;
    if (i >= (long long)N * (HID / 4)) return;
    const int n = (int)(i >> 4);
    const int q = (int)(i & 15);
    const float di = dinv[n];
    const float s = di * di;
    float4 a = agg[i], hwv = hw[i], bv = bias[q];
    float4 v;
    v.x = a.x + s * hwv.x + bv.x;
    v.y = a.y + s * hwv.y + bv.y;
    v.z = a.z + s * hwv.z + bv.z;
    v.w = a.w + s * hwv.w + bv.w;
    if (relu) {
        v.x = fmaxf(v.x, 0.0f); v.y = fmaxf(v.y, 0.0f);
        v.z = fmaxf(v.z, 0.0f); v.w = fmaxf(v.w, 0.0f);
    }
    hout[i] = v;
}

__global__ void pool_accum_kernel(const float4* __restrict__ h,
                                  const int* __restrict__ batch,
                                  float* __restrict__ sums,
                                  float* __restrict__ counts, int N) {
    long long i = (long long)blockIdx.x * blockDim.x + threadIdx.x;
    if (i >= (long long)N * (HID / 4)) return;
    const int n = (int)(i >> 4);
    const int q = (int)(i & 15) << 2;
    const int b = batch[n];
    float4 v = h[i];
    float* sp = sums + (size_t)b * HID + q;
    atomicAdd(sp + 0, v.x);
    atomicAdd(sp + 1, v.y);
    atomicAdd(sp + 2, v.z);
    atomicAdd(sp + 3, v.w);
    if (q == 0) atomicAdd(&counts[b], 1.0f);
}

__global__ void pool_div_kernel(float* __restrict__ sums,
                                const float* __restrict__ counts, int B) {
    int i = blockIdx.x * blockDim.x + threadIdx.x;
    if (i < B * HID) sums[i] /= fmaxf(counts[i >> 6], 1.0f);
}

__global__ void build_z_kernel(const int* __restrict__ state,
                               const float4* __restrict__ emb,
                               const float4* __restrict__ g,
                               float4* __restrict__ z, int B) {
    const int ZV = 3 * EMB / 4;  // float4s per row of z
    int i = blockIdx.x * blockDim.x + threadIdx.x;
    if (i >= B * ZV) return;
    const int b = i / ZV;
    const int c = i - b * ZV;    // float4 column
    float4 v;
    if (c < EMB / 4)            v = emb[(size_t)state[b * 2 + 0] * (EMB / 4) + c];
    else if (c < 2 * (EMB / 4)) v = emb[(size_t)state[b * 2 + 1] * (EMB / 4) + (c - EMB / 4)];
    else                        v = g[(size_t)b * (EMB / 4) + (c - 2 * (EMB / 4))];
    z[i] = v;
}

// ---------------------------------------------------------------------------
// Host-side orchestration
// ---------------------------------------------------------------------------
static void launch_gemm(const float* A, const int* gidx, int lda,
                        const float* B, const float* bias, float* C,
                        int M, int Ncols, int K, int relu, hipStream_t s) {
    const int rowBlocks = ceil_div_i((M + 15) / 16, 8);
    if (Ncols % 64 == 0) {
        dim3 grid(rowBlocks, Ncols / 64);
        wmma_gemm_f32_kernel<4><<<grid, 256, 0, s>>>(A, gidx, lda, B, bias, C, M, Ncols, K, relu);
    } else { // Ncols == 32
        dim3 grid(rowBlocks, Ncols / 32);
        wmma_gemm_f32_kernel<2><<<grid, 256, 0, s>>>(A, gidx, lda, B, bias, C, M, Ncols, K, relu);
    }
}

extern "C" void kernel_launch(void* const* d_in, const int* in_sizes, int n_in,
                              void* d_out, int out_size, void* d_ws, size_t ws_size,
                              hipStream_t stream) {
    const int*   state = (const int*)  d_in[0];
    const int*   x     = (const int*)  d_in[1];
    const int*   eidx  = (const int*)  d_in[2];
    const float* ew    = (const float*)d_in[3];
    const int*   batch = (const int*)  d_in[4];
    const float* emb   = (const float*)d_in[5];
    const float* W1    = (const float*)d_in[6];
    const float* b1    = (const float*)d_in[7];
    const float* W2    = (const float*)d_in[8];
    const float* b2    = (const float*)d_in[9];
    const float* Wfc   = (const float*)d_in[10];
    const float* bfc   = (const float*)d_in[11];
    const float* Wfc1  = (const float*)d_in[12];
    const float* bfc1  = (const float*)d_in[13];
    const float* Wfc2  = (const float*)d_in[14];
    const float* bfc2  = (const float*)d_in[15];
    const float* Wfc3  = (const float*)d_in[16];
    const float* bfc3  = (const float*)d_in[17];

    const int N = in_sizes[1];
    const int E = in_sizes[3];
    const int B = in_sizes[0] / 2;
    const int* esrc = eidx;
    const int* edst = eidx + E;

    // Workspace carve-up (floats); every offset is 16B-aligned for b128 access
    float* ws = (float*)d_ws;
    size_t off = 0;
    float* dinv   = ws + off; off += (size_t)N;            // deg -> dinv in place
    float* nrm    = ws + off; off += (size_t)E;
    float* hw     = ws + off; off += (size_t)N * HID;
    float* agg    = ws + off; off += (size_t)N * HID;
    float* h      = ws + off; off += (size_t)N * HID;      // h1 then h2 (reused)
    float* sums   = ws + off; off += (size_t)B * HID;      // -> pooled in place
    float* counts = ws + off; off += (size_t)B * 4;        // keep 16B alignment
    float* g      = ws + off; off += (size_t)B * EMB;
    float* z      = ws + off; off += (size_t)B * 3 * EMB;
    float* z1     = ws + off; off += (size_t)B * HID;
    float* z2     = ws + off; off += (size_t)B * HID;
    (void)ws_size; (void)n_in;

    const long long NH  = (long long)N * HID;
    const long long NH4 = NH / 4;
    const long long EH4 = (long long)E * 16;

    // Degree + normalization coefficients
    fill_kernel<<<ceil_div_i(N, 256), 256, 0, stream>>>(dinv, 1.0f, N);
    deg_accum_kernel<<<ceil_div_i(E, 256), 256, 0, stream>>>(edst, ew, dinv, E);
    rsqrt_inplace_kernel<<<ceil_div_i(N, 256), 256, 0, stream>>>(dinv, N);
    norm_kernel<<<ceil_div_i(E, 256), 256, 0, stream>>>(esrc, edst, ew, dinv, nrm, E);

    // Conv1: hw = emb_table[x] @ W1  (gathered A rows, K=256)
    launch_gemm(emb, x, EMB, W1, nullptr, hw, N, HID, EMB, 0, stream);
    fill_kernel<<<ceil_div_i(NH, 256), 256, 0, stream>>>(agg, 0.0f, NH);
    edge_scatter_kernel<<<ceil_div_i(EH4, 256), 256, 0, stream>>>(esrc, edst, nrm, hw, agg, EH4);
    combine_kernel<<<ceil_div_i(NH4, 256), 256, 0, stream>>>(
        (const float4*)agg, (const float4*)hw, dinv, (const float4*)b1, (float4*)h, N, 1);

    // Conv2: hw = h @ W2 (K=64), no relu at the end
    launch_gemm(h, nullptr, HID, W2, nullptr, hw, N, HID, HID, 0, stream);
    fill_kernel<<<ceil_div_i(NH, 256), 256, 0, stream>>>(agg, 0.0f, NH);
    edge_scatter_kernel<<<ceil_div_i(EH4, 256), 256, 0, stream>>>(esrc, edst, nrm, hw, agg, EH4);
    combine_kernel<<<ceil_div_i(NH4, 256), 256, 0, stream>>>(
        (const float4*)agg, (const float4*)hw, dinv, (const float4*)b2, (float4*)h, N, 0);

    // Mean pool over batch
    fill_kernel<<<ceil_div_i(B * HID, 256), 256, 0, stream>>>(sums, 0.0f, B * HID);
    fill_kernel<<<1, 256, 0, stream>>>(counts, 0.0f, B);
    pool_accum_kernel<<<ceil_div_i(NH4, 256), 256, 0, stream>>>(
        (const float4*)h, batch, sums, counts, N);
    pool_div_kernel<<<ceil_div_i(B * HID, 256), 256, 0, stream>>>(sums, counts, B);

    // Head: g = pooled @ Wfc + bfc ; z = [emb[s0], emb[s1], g]
    launch_gemm(sums, nullptr, HID, Wfc, bfc, g, B, EMB, HID, 0, stream);
    build_z_kernel<<<ceil_div_i(B * (3 * EMB / 4), 256), 256, 0, stream>>>(
        state, (const float4*)emb, (const float4*)g, (float4*)z, B);
    launch_gemm(z,  nullptr, 3 * EMB, Wfc1, bfc1, z1, B, HID, 3 * EMB, 1, stream);
    launch_gemm(z1, nullptr, HID,     Wfc2, bfc2, z2, B, HID, HID,     1, stream);
    launch_gemm(z2, nullptr, HID,     Wfc3, bfc3, (float*)d_out, B, OUTF, HID, 0, stream);
    (void)out_size;
}